// CausalLocalMultiHeadSelfConvAttention_79018808312520
// MI455X (gfx1250) — compile-verified
//
#include <hip/hip_runtime.h>
#include <hip/hip_bf16.h>
#include <math.h>

// Problem dims (fixed by the reference)
constexpr int B_ = 2;
constexpr int C_ = 512;
constexpr int T_ = 2048;
constexpr int W_ = 32;                 // window length (W+1 = 33 taps)
constexpr int BCT_ = B_ * C_ * T_;     // 2,097,152
constexpr float INV_SCALE = 0.125f;    // 1/sqrt(64)

typedef float v2f __attribute__((ext_vector_type(2)));
typedef float v8f __attribute__((ext_vector_type(8)));

// ---------------------------------------------------------------------------
// Kernel A: causal depthwise conv (K=3) -> q, k/SCALE, v
// ---------------------------------------------------------------------------
__global__ void qkv_conv_kernel(const float* __restrict__ x,
                                const float* __restrict__ wq,
                                const float* __restrict__ wk,
                                const float* __restrict__ wv,
                                float* __restrict__ q,
                                float* __restrict__ k,
                                float* __restrict__ v) {
  int idx = blockIdx.x * blockDim.x + threadIdx.x;
  if (idx >= BCT_) return;
  int t = idx % T_;
  int c = (idx / T_) % C_;
  const float* xr = x + (idx - t);           // x[b,c,:]
  float x0 = (t >= 2) ? xr[t - 2] : 0.f;
  float x1 = (t >= 1) ? xr[t - 1] : 0.f;
  float x2 = xr[t];
  const float* wqc = wq + c * 3;
  const float* wkc = wk + c * 3;
  const float* wvc = wv + c * 3;
  float qv = fmaf(wqc[0], x0, fmaf(wqc[1], x1, wqc[2] * x2));
  float kv = fmaf(wkc[0], x0, fmaf(wkc[1], x1, wkc[2] * x2));
  float vv = fmaf(wvc[0], x0, fmaf(wvc[1], x1, wvc[2] * x2));
  q[idx] = qv;
  k[idx] = kv * INV_SCALE;   // fold 1/sqrt(D) into k
  v[idx] = vv;
}

// ---------------------------------------------------------------------------
// Kernel B: 33-tap masked window softmax, per (b,c,t). Scores stay in VGPRs.
// ---------------------------------------------------------------------------
__global__ void window_softmax_kernel(const float* __restrict__ q,
                                      const float* __restrict__ k,
                                      const float* __restrict__ v,
                                      const int* __restrict__ mask,
                                      float* __restrict__ attn) {
  int idx = blockIdx.x * blockDim.x + threadIdx.x;
  if (idx >= BCT_) return;
  int t = idx % T_;
  int b = idx / (C_ * T_);
  const float* krow = k + (idx - t);
  const float* vrow = v + (idx - t);
  const int* mrow = mask + b * T_;
  float qv = q[idx];

  float s[W_ + 1];
  float mx = -3.0e38f;
#pragma unroll
  for (int w = 0; w <= W_; ++w) {
    int src = t + w - W_;
    int m   = (src >= 0) ? mrow[src] : 0;       // padded mask = 0
    float kv = (src >= 0) ? krow[src] : 0.f;    // padded k = 0
    float sc = (m != 0) ? qv * kv : -1e9f;
    s[w] = sc;
    mx = fmaxf(mx, sc);
  }
  float sum = 0.f;
#pragma unroll
  for (int w = 0; w <= W_; ++w) {
    float p = __expf(s[w] - mx);   // all-masked -> all exp(0)=1 -> uniform, like ref
    s[w] = p;
    sum += p;
  }
  float acc = 0.f;
#pragma unroll
  for (int w = 0; w <= W_; ++w) {
    int src = t + w - W_;
    float vv = (src >= 0) ? vrow[src] : 0.f;    // padded v = 0
    acc = fmaf(s[w], vv, acc);
  }
  attn[idx] = acc / sum;
}

// ---------------------------------------------------------------------------
// Kernel C: out[b,o,t] = sum_c fc_w[o,c]*attn[b,c,t] + fc_b[o]
// M=512, K=512, N=B*T=4096 via V_WMMA_F32_16X16X4_F32 (exact f32).
// 2x2 register blocking: each wave32 produces a 32(M) x 32(N) block
// (4 x v8f accumulators), halving L2 operand traffic vs 1x1 tiles and
// giving 4 WMMAs per 6 loads in the K-loop.
// ---------------------------------------------------------------------------
constexpr int MB_TILES = C_ / 32;               // 16 M-blocks of 32 rows
constexpr int N_TOTAL  = B_ * T_;               // 4096
constexpr int NB_TILES = N_TOTAL / 32;          // 128 N-blocks of 32 cols
constexpr int TOTAL_BLOCKS = MB_TILES * NB_TILES;   // 2048 waves
constexpr int WAVES_PER_BLOCK = 4;

__global__ void fc_wmma_kernel(const float* __restrict__ attn,
                               const float* __restrict__ fc_w,
                               const float* __restrict__ fc_b,
                               float* __restrict__ out) {
  int lane = threadIdx.x & 31;
  int wave = threadIdx.x >> 5;
  int blk  = blockIdx.x * WAVES_PER_BLOCK + wave;
  if (blk >= TOTAL_BLOCKS) return;        // wave-uniform; EXEC stays all-ones
  int mBlk = blk & (MB_TILES - 1);
  int nBlk = blk / MB_TILES;
  int o0 = mBlk * 32;
  int n0 = nBlk * 32;
  int b  = n0 / T_;                       // 32-col blocks never straddle batches
  int t0 = n0 % T_;

  int lhalf = lane & 15;                  // M (for A) / N (for B,C,D) index
  int ksel  = (lane >> 4) << 1;           // lanes 16-31 hold K+2, K+3

  // A rows: fc_w row-major (O x C); two M-tiles, 16 rows apart
  const float* arow0 = fc_w + (o0 + lhalf) * C_ + ksel;        // 8B aligned
  const float* arow1 = arow0 + 16 * C_;
  // B cols: attn[b, c, t]; two N-tiles, 16 cols apart
  const float* bcol0 = attn + b * (C_ * T_) + t0 + lhalf;
  const float* bcol1 = bcol0 + 16;

  v8f acc00 = {0.f,0.f,0.f,0.f,0.f,0.f,0.f,0.f};
  v8f acc01 = acc00, acc10 = acc00, acc11 = acc00;

#pragma unroll 4
  for (int kk = 0; kk < C_; kk += 4) {
    v2f a0 = *(const v2f*)(arow0 + kk);
    v2f a1 = *(const v2f*)(arow1 + kk);
    v2f b0, b1;
    int r0 = (kk + ksel) * T_;
    int r1 = (kk + ksel + 1) * T_;
    b0.x = bcol0[r0];  b0.y = bcol0[r1];
    b1.x = bcol1[r0];  b1.y = bcol1[r1];
    // 8 args: (neg_a, A, neg_b, B, c_mod, C, reuse_a, reuse_b)
    acc00 = __builtin_amdgcn_wmma_f32_16x16x4_f32(false, a0, false, b0, (short)0, acc00, false, false);
    acc01 = __builtin_amdgcn_wmma_f32_16x16x4_f32(false, a0, false, b1, (short)0, acc01, false, false);
    acc10 = __builtin_amdgcn_wmma_f32_16x16x4_f32(false, a1, false, b0, (short)0, acc10, false, false);
    acc11 = __builtin_amdgcn_wmma_f32_16x16x4_f32(false, a1, false, b1, (short)0, acc11, false, false);
  }

  // C/D layout: VGPR r holds rows M=r (lanes 0-15) and M=r+8 (lanes 16-31)
  float* obase0 = out + b * (C_ * T_) + t0 + lhalf;        // N-tile 0
  float* obase1 = obase0 + 16;                             // N-tile 1
  int mAdd = (lane >> 4) << 3;
#pragma unroll
  for (int r = 0; r < 8; ++r) {
    int oa = o0 + mAdd + r;           // M-tile 0 rows
    int ob = oa + 16;                 // M-tile 1 rows
    float ba = fc_b[oa];
    float bb = fc_b[ob];
    obase0[oa * T_] = acc00[r] + ba;
    obase1[oa * T_] = acc01[r] + ba;
    obase0[ob * T_] = acc10[r] + bb;
    obase1[ob * T_] = acc11[r] + bb;
  }
}

// ---------------------------------------------------------------------------
// Kernel D: second tuple output = mask passthrough (int32 bits)
// ---------------------------------------------------------------------------
__global__ void mask_copy_kernel(const int* __restrict__ mask,
                                 int* __restrict__ outm) {
  int i = blockIdx.x * blockDim.x + threadIdx.x;
  if (i < B_ * T_) outm[i] = mask[i];
}

extern "C" void kernel_launch(void* const* d_in, const int* in_sizes, int n_in,
                              void* d_out, int out_size, void* d_ws, size_t ws_size,
                              hipStream_t stream) {
  const float* x    = (const float*)d_in[0];
  const int*   mask = (const int*)d_in[1];
  const float* wq   = (const float*)d_in[2];
  const float* wk   = (const float*)d_in[3];
  const float* wv   = (const float*)d_in[4];
  const float* fc_w = (const float*)d_in[5];
  const float* fc_b = (const float*)d_in[6];
  float* out = (float*)d_out;

  // Workspace: q, k(prescaled), v, attn  -> 4 * 8 MB = 32 MB
  float* qws = (float*)d_ws;
  float* kws = qws + BCT_;
  float* vws = kws + BCT_;
  float* aws = vws + BCT_;

  const int threads = 256;
  const int blocksA = (BCT_ + threads - 1) / threads;
  qkv_conv_kernel<<<blocksA, threads, 0, stream>>>(x, wq, wk, wv, qws, kws, vws);
  window_softmax_kernel<<<blocksA, threads, 0, stream>>>(qws, kws, vws, mask, aws);
  fc_wmma_kernel<<<TOTAL_BLOCKS / WAVES_PER_BLOCK, WAVES_PER_BLOCK * 32, 0, stream>>>(
      aws, fc_w, fc_b, out);
  mask_copy_kernel<<<(B_ * T_ + 255) / 256, 256, 0, stream>>>(
      mask, (int*)(out + BCT_));
}